// PAU_conv_71330816852553
// MI455X (gfx1250) — compile-verified
//
#include <hip/hip_runtime.h>

// PAU (Pade [5/4], version 'A') elementwise activation, per-channel coeffs.
// x: [B,C,H,W] f32.  Bandwidth-bound: 205 MB total traffic, ~8.8us floor @23.3TB/s.
// One block per (b,c) plane -> channel is wave-uniform -> coeffs live in SGPRs.

typedef __attribute__((ext_vector_type(4))) float v4f;

constexpr int Bc = 32, Cc = 256, Hc = 56, Wc = 56;
constexpr int HW  = Hc * Wc;   // 3136 floats per plane
constexpr int HW4 = HW / 4;    // 784 float4 per plane
constexpr int NPLANES = Bc * Cc;
constexpr int TPB = 256;       // 8 waves (wave32)

__device__ __forceinline__ v4f pade54(v4f v, float cc,
                                      float a0, float a1, float a2,
                                      float a3, float a4, float a5,
                                      float b1, float b2, float b3, float b4) {
    v4f r;
#pragma unroll
    for (int k = 0; k < 4; ++k) {
        float y = v[k] + cc;
        // P(y) = a0 + a1 y + ... + a5 y^5  (Horner, 5 FMA)
        float p = fmaf(fmaf(fmaf(fmaf(fmaf(a5, y, a4), y, a3), y, a2), y, a1), y, a0);
        // Q(y) = 1 + |y * (b1 + b2 y + b3 y^2 + b4 y^3)|  (3 FMA + mul/abs/add)
        float q = fmaf(fmaf(fmaf(b4, y, b3), y, b2), y, b1);
        q = 1.0f + __builtin_fabsf(y * q);
        // q >= 1 always -> v_rcp_f32 + one Newton step is safe & ~1 ulp
        float rq = __builtin_amdgcn_rcpf(q);
        rq = rq * fmaf(-q, rq, 2.0f);
        r[k] = p * rq;
    }
    return r;
}

__global__ __launch_bounds__(TPB, 4)
void pau_plane_kernel(const float* __restrict__ x,
                      const float* __restrict__ num,   // [C,6]
                      const float* __restrict__ den,   // [C,4]
                      const float* __restrict__ ctr,   // [C]
                      float* __restrict__ out) {
    const int plane = blockIdx.x;          // b*C + c
    const int c     = plane & (Cc - 1);    // C = 256, power of two

    // Wave-uniform addresses -> scalar loads into SGPRs.
    const float a0 = num[c * 6 + 0], a1 = num[c * 6 + 1], a2 = num[c * 6 + 2];
    const float a3 = num[c * 6 + 3], a4 = num[c * 6 + 4], a5 = num[c * 6 + 5];
    const float b1 = den[c * 4 + 0], b2 = den[c * 4 + 1];
    const float b3 = den[c * 4 + 2], b4 = den[c * 4 + 3];
    const float cc = ctr[c];

    const v4f* __restrict__ xin  = (const v4f*)(x   + (size_t)plane * HW);
    v4f*       __restrict__ yout = (v4f*)      (out + (size_t)plane * HW);

    const int t = threadIdx.x;

    // gfx1250 prefetch: pull the next plane's lines toward L2 while we compute.
    __builtin_prefetch(xin + HW4 + t, 0, 1);

    // 784 = 3*256 + 16: three full vec4 rounds + small tail.
    // Issue all loads first (NT: streamed once, don't thrash 192MB L2).
    v4f v0 = __builtin_nontemporal_load(xin + t);
    v4f v1 = __builtin_nontemporal_load(xin + t + TPB);
    v4f v2 = __builtin_nontemporal_load(xin + t + 2 * TPB);
    const bool tail = t < (HW4 - 3 * TPB);   // t < 16
    v4f v3;
    if (tail) v3 = __builtin_nontemporal_load(xin + t + 3 * TPB);

    v4f r0 = pade54(v0, cc, a0, a1, a2, a3, a4, a5, b1, b2, b3, b4);
    v4f r1 = pade54(v1, cc, a0, a1, a2, a3, a4, a5, b1, b2, b3, b4);
    v4f r2 = pade54(v2, cc, a0, a1, a2, a3, a4, a5, b1, b2, b3, b4);

    __builtin_nontemporal_store(r0, yout + t);
    __builtin_nontemporal_store(r1, yout + t + TPB);
    __builtin_nontemporal_store(r2, yout + t + 2 * TPB);
    if (tail) {
        v4f r3 = pade54(v3, cc, a0, a1, a2, a3, a4, a5, b1, b2, b3, b4);
        __builtin_nontemporal_store(r3, yout + t + 3 * TPB);
    }
}

extern "C" void kernel_launch(void* const* d_in, const int* in_sizes, int n_in,
                              void* d_out, int out_size, void* d_ws, size_t ws_size,
                              hipStream_t stream) {
    const float* x   = (const float*)d_in[0];  // [B,C,H,W]
    const float* num = (const float*)d_in[1];  // [C,6]
    const float* den = (const float*)d_in[2];  // [C,4]
    const float* ctr = (const float*)d_in[3];  // [C]
    float* out = (float*)d_out;

    pau_plane_kernel<<<dim3(NPLANES), dim3(TPB), 0, stream>>>(x, num, den, ctr, out);
}